// GlobalWeBGNN_32822140076794
// MI455X (gfx1250) — compile-verified
//
#include <hip/hip_runtime.h>
#include <math.h>

typedef __attribute__((ext_vector_type(2))) float v2f;
typedef __attribute__((ext_vector_type(8))) float v8f;

#define N_NODES 50000
#define N_EDGES 800000

// ---------------------------------------------------------------------------
// Utility kernels
// ---------------------------------------------------------------------------
__global__ void zero_kernel(float* __restrict__ p, int n) {
    int i = blockIdx.x * blockDim.x + threadIdx.x;
    if (i < n) p[i] = 0.0f;
}

// zero the up/dn portion (cols 0..127) of the cat buffer [N,192]
__global__ void zero_cat_updn(float* __restrict__ cat, int n_nodes) {
    int i = blockIdx.x * blockDim.x + threadIdx.x;
    int node = i >> 7;
    int f = i & 127;
    if (node < n_nodes) cat[(size_t)node * 192 + f] = 0.0f;
}

// deg_up[col[e]] += 1 ; deg_dn[row[e]] += 1
__global__ void degree_kernel(const long long* __restrict__ row,
                              const long long* __restrict__ col,
                              float* __restrict__ deg_up,
                              float* __restrict__ deg_dn, int E) {
    int e = blockIdx.x * blockDim.x + threadIdx.x;
    if (e >= E) return;
    atomicAdd(&deg_up[(int)col[e]], 1.0f);
    atomicAdd(&deg_dn[(int)row[e]], 1.0f);
}

// in-place: d -> (d > 0 ? 1/d : 0), over 2N contiguous floats
__global__ void invert_kernel(float* __restrict__ d, int n) {
    int i = blockIdx.x * blockDim.x + threadIdx.x;
    if (i >= n) return;
    float v = d[i];
    d[i] = (v > 0.0f) ? (1.0f / v) : 0.0f;
}

// ---------------------------------------------------------------------------
// WMMA fp32 GEMM:  Y[m, colOff + n] = sum_k X[m,k] * W[k,n],  n in [0,64)
// one block = 128 threads = 4 waves; block handles 16 rows x 64 cols;
// wave w handles the 16x16 tile at columns [16w, 16w+16).
// K is 64 (layer 1) or 192 (layers 2,3).  N_NODES % 16 == 0.
// ---------------------------------------------------------------------------
__global__ void __launch_bounds__(128)
gemm_wmma_f32(const float* __restrict__ X, const float* __restrict__ W,
              float* __restrict__ Y, int K, int ldY, int colOff) {
    __shared__ float Xt[16 * 192];  // max K = 192 -> 12 KB
    const int m0 = blockIdx.x * 16;

    // cooperative stage of the 16 x K input tile into LDS
    for (int i = threadIdx.x; i < 16 * K; i += 128) {
        int r = i / K;
        int k = i - r * K;
        Xt[r * K + k] = X[(size_t)(m0 + r) * K + k];
    }
    __syncthreads();

    const int wave = threadIdx.x >> 5;   // 0..3 -> output column tile
    const int lane = threadIdx.x & 31;
    const int half = lane >> 4;          // 0: lanes 0-15, 1: lanes 16-31
    const int l    = lane & 15;
    const int n0   = wave * 16;

    v8f acc = {};
    for (int kb = 0; kb < K; kb += 4) {
        // A fragment 16x4 (f32): lane l holds row M=l; VGPR0/1 hold
        // K = kb+2*half, kb+2*half+1
        v2f a, b;
        a.x = Xt[l * K + kb + 2 * half];
        a.y = Xt[l * K + kb + 2 * half + 1];
        // B fragment 4x16 (f32): lane l holds col N=n0+l, rows mirrored
        b.x = W[(size_t)(kb + 2 * half) * 64 + n0 + l];
        b.y = W[(size_t)(kb + 2 * half + 1) * 64 + n0 + l];
        acc = __builtin_amdgcn_wmma_f32_16x16x4_f32(
            /*neg_a=*/false, a, /*neg_b=*/false, b,
            /*c_mod=*/(short)0, acc, /*reuse_a=*/false, /*reuse_b=*/false);
    }

    // C/D layout: VGPR vi holds M = vi + 8*half, N = n0 + l
#pragma unroll
    for (int vi = 0; vi < 8; ++vi) {
        int m = m0 + vi + 8 * half;
        Y[(size_t)m * ldY + colOff + n0 + l] = acc[vi];
    }
}

// ---------------------------------------------------------------------------
// Edge scatter: one thread per (edge, feature-of-64)
//   cat[col][f]      += inv_up[col] * ewu[e] * up_x[row][f]
//   cat[row][64 + f] += inv_dn[row] * ewd[e] * dn_x[col][f]
// ---------------------------------------------------------------------------
__global__ void __launch_bounds__(256)
scatter_kernel(const float* __restrict__ up_x, const float* __restrict__ dn_x,
               const long long* __restrict__ row, const long long* __restrict__ col,
               const float* __restrict__ ewu, const float* __restrict__ ewd,
               const float* __restrict__ inv_up, const float* __restrict__ inv_dn,
               float* __restrict__ cat, int E) {
    long long tid = (long long)blockIdx.x * blockDim.x + threadIdx.x;
    int e = (int)(tid >> 6);
    int f = (int)(tid & 63);
    if (e >= E) return;
    int r = (int)row[e];
    int c = (int)col[e];
    float cu = inv_up[c] * ewu[e];
    float cd = inv_dn[r] * ewd[e];
    atomicAdd(&cat[(size_t)c * 192 + f],      cu * up_x[(size_t)r * 64 + f]);
    atomicAdd(&cat[(size_t)r * 192 + 64 + f], cd * dn_x[(size_t)c * 64 + f]);
}

// ---------------------------------------------------------------------------
// Per-node L2 normalize (p=2, eps 1e-12) + leaky_relu(0.1)
// one wave32 per node (192 = 32 lanes x 6)
// ---------------------------------------------------------------------------
__global__ void __launch_bounds__(256)
norm_lrelu_kernel(const float* __restrict__ cat, float* __restrict__ out, int n_nodes) {
    int gw = (blockIdx.x * blockDim.x + threadIdx.x) >> 5;  // global wave id = node
    int lane = threadIdx.x & 31;
    if (gw >= n_nodes) return;
    const float* p = cat + (size_t)gw * 192;
    float v[6];
    float ss = 0.0f;
#pragma unroll
    for (int i = 0; i < 6; ++i) {
        v[i] = p[lane + 32 * i];
        ss += v[i] * v[i];
    }
#pragma unroll
    for (int m = 16; m >= 1; m >>= 1) ss += __shfl_xor(ss, m, 32);
    float rn = 1.0f / fmaxf(sqrtf(ss), 1e-12f);
    float* q = out + (size_t)gw * 192;
#pragma unroll
    for (int i = 0; i < 6; ++i) {
        float y = v[i] * rn;
        q[lane + 32 * i] = (y > 0.0f) ? y : 0.1f * y;
    }
}

// mean of the three edge-weight vectors
__global__ void mean_kernel(const float* __restrict__ a, const float* __restrict__ b,
                            const float* __restrict__ c, float* __restrict__ o, int E) {
    int e = blockIdx.x * blockDim.x + threadIdx.x;
    if (e < E) o[e] = (a[e] + b[e] + c[e]) * (1.0f / 3.0f);
}

// ---------------------------------------------------------------------------
// Host launcher
// ---------------------------------------------------------------------------
extern "C" void kernel_launch(void* const* d_in, const int* in_sizes, int n_in,
                              void* d_out, int out_size, void* d_ws, size_t ws_size,
                              hipStream_t stream) {
    const int N = N_NODES, E = N_EDGES;

    const float*     x  = (const float*)d_in[0];
    const long long* ei = (const long long*)d_in[1];   // int64 [2, E]
    const long long* row = ei;
    const long long* col = ei + E;

    const float* W_up[3]   = { (const float*)d_in[2],  (const float*)d_in[7],  (const float*)d_in[12] };
    const float* W_down[3] = { (const float*)d_in[3],  (const float*)d_in[8],  (const float*)d_in[13] };
    const float* W_bias[3] = { (const float*)d_in[4],  (const float*)d_in[9],  (const float*)d_in[14] };
    const float* ew_up[3]  = { (const float*)d_in[5],  (const float*)d_in[10], (const float*)d_in[15] };
    const float* ew_dn[3]  = { (const float*)d_in[6],  (const float*)d_in[11], (const float*)d_in[16] };

    // workspace layout (floats)
    float* ws = (float*)d_ws;
    float* inv_up = ws;                       // [N]   (deg_up then 1/deg_up)
    float* inv_dn = inv_up + N;               // [N]   (contiguous with inv_up)
    float* up_x   = inv_dn + N;               // [N,64]
    float* dn_x   = up_x + (size_t)N * 64;    // [N,64]
    float* cat    = dn_x + (size_t)N * 64;    // [N,192]
    float* hbuf   = cat + (size_t)N * 192;    // [N,192]

    float* out_h  = (float*)d_out;                        // [N,192]
    float* out_mu = out_h + (size_t)N * 192;              // [E]
    float* out_md = out_mu + E;                           // [E]

    // --- degrees & reciprocals (shared by all 3 layers) ---
    zero_kernel<<<(2 * N + 255) / 256, 256, 0, stream>>>(inv_up, 2 * N);
    degree_kernel<<<(E + 255) / 256, 256, 0, stream>>>(row, col, inv_up, inv_dn, E);
    invert_kernel<<<(2 * N + 255) / 256, 256, 0, stream>>>(inv_up, 2 * N);

    const int row_tiles = N / 16;  // 3125 exactly

    for (int l = 0; l < 3; ++l) {
        const float* Xin = (l == 0) ? x : hbuf;
        const int K = (l == 0) ? 64 : 192;

        // three dense GEMMs via fp32 WMMA
        gemm_wmma_f32<<<row_tiles, 128, 0, stream>>>(Xin, W_up[l],   up_x, K, 64,  0);
        gemm_wmma_f32<<<row_tiles, 128, 0, stream>>>(Xin, W_down[l], dn_x, K, 64,  0);
        gemm_wmma_f32<<<row_tiles, 128, 0, stream>>>(Xin, W_bias[l], cat,  K, 192, 128);

        // clear scatter-accumulation region (cols 0..127 of cat)
        zero_cat_updn<<<(N * 128 + 255) / 256, 256, 0, stream>>>(cat, N);

        // edge message passing (both directions)
        long long sthreads = (long long)E * 64;
        scatter_kernel<<<(int)((sthreads + 255) / 256), 256, 0, stream>>>(
            up_x, dn_x, row, col, ew_up[l], ew_dn[l], inv_up, inv_dn, cat, E);

        // L2 normalize + leaky relu -> next-layer input (or final output)
        float* dst = (l == 2) ? out_h : hbuf;
        norm_lrelu_kernel<<<(N * 32 + 255) / 256, 256, 0, stream>>>(cat, dst, N);
    }

    // edge-weight means
    mean_kernel<<<(E + 255) / 256, 256, 0, stream>>>(ew_up[0], ew_up[1], ew_up[2], out_mu, E);
    mean_kernel<<<(E + 255) / 256, 256, 0, stream>>>(ew_dn[0], ew_dn[1], ew_dn[2], out_md, E);
}